// PIEM_33174327394746
// MI455X (gfx1250) — compile-verified
//
#include <hip/hip_runtime.h>
#include <hip/hip_bf16.h>
#include <math.h>

typedef float v2f __attribute__((ext_vector_type(2)));
typedef float v8f __attribute__((ext_vector_type(8)));

#define BB 100
#define C1 512
#define C2 640
#define DD 5
#define SL 21
#define S1 42
#define NN 3200          // C2*DD
#define P1 1764          // S1*S1
#define PL 441           // SL*SL
#define GN_CH 32         // channels per group (C1/16)
#define GLEN 672         // 32*21 elements per group-norm group
#define MPAD 112         // 100 rows padded to 7*16

// ---------------- kernel 1: row/col means of F_L ----------------
__global__ void k_reduce_FL(const float* __restrict__ FL,
                            float* __restrict__ mh, float* __restrict__ mw) {
    __shared__ float s[PL];
    const int bc = blockIdx.x;                 // b*C1 + c
    const float* src = FL + (size_t)bc * PL;
    for (int i = threadIdx.x; i < PL; i += 64) s[i] = src[i];
    __syncthreads();
    if (threadIdx.x < SL) {                    // mean over w -> mh[b,c,h]
        int h = threadIdx.x; float a = 0.f;
        for (int w = 0; w < SL; ++w) a += s[h * SL + w];
        mh[(size_t)bc * SL + h] = a * (1.0f / SL);
    }
    if (threadIdx.x >= 32 && threadIdx.x < 32 + SL) {  // mean over h -> mw[b,c,w]
        int w = threadIdx.x - 32; float a = 0.f;
        for (int h = 0; h < SL; ++h) a += s[h * SL + w];
        mw[(size_t)bc * SL + w] = a * (1.0f / SL);
    }
}

// ---------------- kernel 2: dwconv7 + GroupNorm + sigmoid ----------------
__global__ void k_gate(const float* __restrict__ z, float* __restrict__ xout,
                       const float* __restrict__ scw,
                       const float* __restrict__ gng, const float* __restrict__ gnb) {
    __shared__ float zin[GLEN];
    __shared__ float yc[GLEN];
    __shared__ float red[256];
    __shared__ float red2[256];
    const int b = blockIdx.x >> 4;
    const int g = blockIdx.x & 15;
    const int c0 = g * GN_CH;
    const int tid = threadIdx.x;

    for (int i = tid; i < GLEN; i += 256) {
        int c = i / SL, t = i % SL;
        zin[i] = z[((size_t)(b * C1 + c0 + c)) * SL + t];
    }
    __syncthreads();

    float ls = 0.f, ls2 = 0.f;
    for (int i = tid; i < GLEN; i += 256) {
        int c = i / SL, t = i % SL;
        float acc = 0.f;
        #pragma unroll
        for (int k = 0; k < 7; ++k) {
            int tt = t + k - 3;
            if (tt >= 0 && tt < SL) acc += scw[(c0 + c) * 7 + k] * zin[c * SL + tt];
        }
        yc[i] = acc; ls += acc; ls2 += acc * acc;
    }
    red[tid] = ls; red2[tid] = ls2;
    __syncthreads();
    for (int s = 128; s > 0; s >>= 1) {
        if (tid < s) { red[tid] += red[tid + s]; red2[tid] += red2[tid + s]; }
        __syncthreads();
    }
    const float mean = red[0] * (1.0f / GLEN);
    const float var  = red2[0] * (1.0f / GLEN) - mean * mean;
    const float rs   = rsqrtf(var + 1e-5f);
    for (int i = tid; i < GLEN; i += 256) {
        int c = i / SL, t = i % SL;
        float xn = (yc[i] - mean) * rs;
        float v  = xn * gng[c0 + c] + gnb[c0 + c];
        xout[((size_t)(b * C1 + c0 + c)) * SL + t] = 1.0f / (1.0f + expf(-v));
    }
}

// ---------------- kernel 3: t16[b,h,w,o] = sum_c mb1[o,c]*FL*xh*xw ----------------
__global__ void k_t16(const float* __restrict__ FL,
                      const float* __restrict__ xh, const float* __restrict__ xw,
                      const float* __restrict__ mb1w, float* __restrict__ t16) {
    const int b = blockIdx.x / SL;
    const int h = blockIdx.x % SL;
    const int tid = threadIdx.x;
    if (tid >= SL * 16) return;
    const int w = tid >> 4;
    const int o = tid & 15;
    float acc = 0.f;
    for (int c = 0; c < C1; ++c) {
        size_t bc = (size_t)(b * C1 + c);
        float f = FL[(bc * SL + h) * SL + w];
        float a = xh[bc * SL + h];
        float d = xw[bc * SL + w];
        acc += f * a * d * mb1w[o * C1 + c];
    }
    t16[(((size_t)b * SL + h) * SL + w) * 16 + o] = acc;
}

// ---------------- kernel 4: bilinear 21->42 + BN/ReLU + 16->5 + sigmoid ----------------
__global__ void k_mask(const float* __restrict__ t16,
                       const float* __restrict__ mb1b,
                       const float* __restrict__ bng, const float* __restrict__ bnb,
                       const float* __restrict__ mb2w, const float* __restrict__ mb2b,
                       float* __restrict__ mask) {
    int idx = blockIdx.x * blockDim.x + threadIdx.x;
    if (idx >= BB * P1) return;
    const int b = idx / P1;
    const int r = idx % P1;
    const int H = r / S1, W = r % S1;

    float ic = 0.5f * H - 0.25f;
    int i0 = (int)floorf(ic); float fh = ic - (float)i0;
    int i1 = i0 + 1;
    i0 = min(max(i0, 0), SL - 1); i1 = min(max(i1, 0), SL - 1);
    float jc = 0.5f * W - 0.25f;
    int j0 = (int)floorf(jc); float fw = jc - (float)j0;
    int j1 = j0 + 1;
    j0 = min(max(j0, 0), SL - 1); j1 = min(max(j1, 0), SL - 1);

    const float* t00 = t16 + (((size_t)b * SL + i0) * SL + j0) * 16;
    const float* t01 = t16 + (((size_t)b * SL + i0) * SL + j1) * 16;
    const float* t10 = t16 + (((size_t)b * SL + i1) * SL + j0) * 16;
    const float* t11 = t16 + (((size_t)b * SL + i1) * SL + j1) * 16;
    const float w00 = (1.f - fh) * (1.f - fw), w01 = (1.f - fh) * fw;
    const float w10 = fh * (1.f - fw),         w11 = fh * fw;
    const float inv = rsqrtf(1.0f + 1e-5f);

    float hval[16];
    #pragma unroll
    for (int o = 0; o < 16; ++o) {
        float hv = w00 * t00[o] + w01 * t01[o] + w10 * t10[o] + w11 * t11[o];
        hv += mb1b[o];
        hv = hv * (bng[o] * inv) + bnb[o];
        hval[o] = fmaxf(hv, 0.f);
    }
    #pragma unroll
    for (int d = 0; d < DD; ++d) {
        float heat = mb2b[d];
        #pragma unroll
        for (int o = 0; o < 16; ++o) heat += mb2w[d * 16 + o] * hval[o];
        mask[((size_t)(b * DD + d)) * P1 + r] = 1.0f / (1.0f + expf(-heat));
    }
}

// ---------------- kernel 5: sow[b,d] = sum_p mask + 1e-5 ----------------
__global__ void k_sow(const float* __restrict__ mask, float* __restrict__ sow) {
    __shared__ float red[256];
    const int bd = blockIdx.x;
    float a = 0.f;
    for (int p = threadIdx.x; p < P1; p += 256) a += mask[(size_t)bd * P1 + p];
    red[threadIdx.x] = a;
    __syncthreads();
    for (int s = 128; s > 0; s >>= 1) {
        if (threadIdx.x < s) red[threadIdx.x] += red[threadIdx.x + s];
        __syncthreads();
    }
    if (threadIdx.x == 0) sow[bd] = red[0] + 1e-5f;
}

// ---------------- kernel 6: vec + bn1d (one wave per (b,c)) ----------------
__global__ void k_vec(const float* __restrict__ F1, const float* __restrict__ mask,
                      const float* __restrict__ sow,
                      const float* __restrict__ g1d, const float* __restrict__ b1d,
                      float* __restrict__ vec, float* __restrict__ xpad) {
    const int widx = blockIdx.x * (blockDim.x / 32) + (threadIdx.x >> 5);
    const int lane = threadIdx.x & 31;
    const int b = widx / C2, c = widx % C2;
    const float* f1 = F1 + (size_t)(b * C2 + c) * P1;
    const float* mk = mask + (size_t)b * DD * P1;
    float acc[DD] = {0.f, 0.f, 0.f, 0.f, 0.f};
    for (int p = lane; p < P1; p += 32) {
        float f = f1[p];
        #pragma unroll
        for (int d = 0; d < DD; ++d) acc[d] += f * mk[d * P1 + p];
    }
    #pragma unroll
    for (int d = 0; d < DD; ++d) {
        #pragma unroll
        for (int off = 16; off > 0; off >>= 1)
            acc[d] += __shfl_down(acc[d], off, 32);
    }
    if (lane == 0) {
        const float inv = rsqrtf(1.0f + 1e-5f);
        #pragma unroll
        for (int d = 0; d < DD; ++d) {
            int n = c * DD + d;
            float v = acc[d] / sow[b * DD + d];
            vec[(size_t)b * NN + n] = v;
            xpad[(size_t)b * NN + n] = v * (g1d[n] * inv) + b1d[n];
        }
    }
}

// ---------------- kernel 6b: zero pad rows 100..111 of xpad ----------------
__global__ void k_padzero(float* __restrict__ xpad) {
    int i = blockIdx.x * blockDim.x + threadIdx.x;
    if (i < (MPAD - BB) * NN) xpad[(size_t)BB * NN + i] = 0.f;
}

// ---------------- kernel 7: WMMA f32 GEMM  z = elu(x @ linW^T + b), blend ----------------
__global__ void k_gemm(const float* __restrict__ xpad, const float* __restrict__ linw,
                       const float* __restrict__ linb, const float* __restrict__ vec,
                       float* __restrict__ zfull) {
    const int n0 = blockIdx.x * 16;
    const int m0 = blockIdx.y * 16;
    const int lane = threadIdx.x;
    const int lm = lane & 15;
    const int lh = lane >> 4;

    const float* Ab = xpad + (size_t)(m0 + lm) * NN + 2 * lh;   // A: rows of x
    const float* Bb = linw + (size_t)(n0 + lm) * NN + 2 * lh;   // B: rows of lin_w (= cols of W^T)

    v8f acc = {};
    #pragma unroll 4
    for (int k = 0; k < NN; k += 4) {
        v2f a = *(const v2f*)(Ab + k);
        v2f b = *(const v2f*)(Bb + k);
        acc = __builtin_amdgcn_wmma_f32_16x16x4_f32(
            false, a, false, b, (short)0, acc, false, false);
    }
    const int n = n0 + lm;
    const float bias = linb[n];
    #pragma unroll
    for (int r = 0; r < 8; ++r) {
        int m = m0 + r + 8 * lh;
        if (m < BB) {
            float zl = acc[r] + bias;
            float e = (zl > 0.f) ? zl : expm1f(zl);
            zfull[(size_t)m * NN + n] = 0.5f * e + 0.5f * vec[(size_t)m * NN + n];
        }
    }
}

// ---------------- kernel 8: support mean + query copy ----------------
__global__ void k_final(const float* __restrict__ zfull, float* __restrict__ out) {
    int i = blockIdx.x * blockDim.x + threadIdx.x;
    const int total = 80 * NN;
    if (i >= total) return;
    if (i < 5 * NN) {
        int w = i / NN, n = i % NN;
        float a = 0.f;
        #pragma unroll
        for (int s = 0; s < 5; ++s) a += zfull[(size_t)(w * 5 + s) * NN + n];
        out[i] = a * 0.2f;
    } else {
        int j = i - 5 * NN;
        int row = 25 + j / NN, n = j % NN;
        out[i] = zfull[(size_t)row * NN + n];
    }
}

extern "C" void kernel_launch(void* const* d_in, const int* in_sizes, int n_in,
                              void* d_out, int out_size, void* d_ws, size_t ws_size,
                              hipStream_t stream) {
    const float* F1   = (const float*)d_in[0];
    const float* F_L  = (const float*)d_in[1];
    const float* scw  = (const float*)d_in[2];
    const float* gng  = (const float*)d_in[3];
    const float* gnb  = (const float*)d_in[4];
    const float* mb1w = (const float*)d_in[5];
    const float* mb1b = (const float*)d_in[6];
    const float* bng  = (const float*)d_in[7];
    const float* bnb  = (const float*)d_in[8];
    const float* mb2w = (const float*)d_in[9];
    const float* mb2b = (const float*)d_in[10];
    const float* g1d  = (const float*)d_in[11];
    const float* b1d  = (const float*)d_in[12];
    const float* linw = (const float*)d_in[13];
    const float* linb = (const float*)d_in[14];
    float* out = (float*)d_out;

    float* ws = (float*)d_ws;
    const size_t MH = (size_t)BB * C1 * SL;             // 1,075,200
    float* mh   = ws;               float* mw   = mh + MH;
    float* xh   = mw + MH;          float* xw   = xh + MH;
    float* t16  = xw + MH;                              // 705,600
    float* mask = t16 + (size_t)BB * 16 * PL;           // 882,000
    float* sow  = mask + (size_t)BB * DD * P1;          // 512 (padded)
    float* vec  = sow + 512;                            // 320,000
    float* xpad = vec + (size_t)BB * NN;                // 358,400
    float* zful = xpad + (size_t)MPAD * NN;             // 320,000

    k_reduce_FL<<<dim3(BB * C1), dim3(64), 0, stream>>>(F_L, mh, mw);
    k_gate<<<dim3(BB * 16), dim3(256), 0, stream>>>(mh, xh, scw, gng, gnb);
    k_gate<<<dim3(BB * 16), dim3(256), 0, stream>>>(mw, xw, scw, gng, gnb);
    k_t16<<<dim3(BB * SL), dim3(352), 0, stream>>>(F_L, xh, xw, mb1w, t16);
    k_mask<<<dim3((BB * P1 + 255) / 256), dim3(256), 0, stream>>>(
        t16, mb1b, bng, bnb, mb2w, mb2b, mask);
    k_sow<<<dim3(BB * DD), dim3(256), 0, stream>>>(mask, sow);
    k_vec<<<dim3(BB * C2 / 8), dim3(256), 0, stream>>>(F1, mask, sow, g1d, b1d, vec, xpad);
    k_padzero<<<dim3(((MPAD - BB) * NN + 255) / 256), dim3(256), 0, stream>>>(xpad);
    k_gemm<<<dim3(NN / 16, MPAD / 16), dim3(32), 0, stream>>>(xpad, linw, linb, vec, zful);
    k_final<<<dim3((80 * NN + 255) / 256), dim3(256), 0, stream>>>(zful, out);
}